// GATConv_40716289966350
// MI455X (gfx1250) — compile-verified
//
#include <hip/hip_runtime.h>
#include <hip/hip_bf16.h>

// GATConv forward for MI455X (gfx1250, wave32).
// Phase 1: H = X@W + b via V_WMMA_F32_16X16X4_F32, fused s_src/s_dst epilogue.
// Phase 2: edge scores -> segment max -> segment sum (float atomics).
// Phase 3: weighted scatter-add aggregation + ELU.

typedef float v2f __attribute__((ext_vector_type(2)));
typedef float v8f __attribute__((ext_vector_type(8)));

#define DIN 128
#define DOUT 64
#define WAVES 4
#define XS_STRIDE 129   // 128 + 1 pad -> conflict-free column reads

__device__ __forceinline__ float elu1(float x) {
    return x > 0.0f ? x : (expf(x) - 1.0f);
}

// Monotone mapping float -> uint so integer atomicMax implements float max.
__device__ __forceinline__ unsigned enc_f32(float x) {
    unsigned u = __float_as_uint(x);
    return (u & 0x80000000u) ? ~u : (u | 0x80000000u);
}
__device__ __forceinline__ float dec_f32(unsigned e) {
    return __uint_as_float((e & 0x80000000u) ? (e ^ 0x80000000u) : ~e);
}

// ---------------------------------------------------------------------------
// Kernel 1: H = X@W + b  (WMMA f32 16x16x4), fused s_src = H@a_src, s_dst = H@a_dst
// One wave per 16-row stripe; 4 column tiles of 16 cover DOUT=64.
// ---------------------------------------------------------------------------
__global__ __launch_bounds__(32 * WAVES) void gat_gemm(
    const float* __restrict__ X, const float* __restrict__ W,
    const float* __restrict__ bias,
    const float* __restrict__ a_src, const float* __restrict__ a_dst,
    float* __restrict__ H, float* __restrict__ ssrc, float* __restrict__ sdst,
    int N)
{
    __shared__ float xs[WAVES][16 * XS_STRIDE];
    const int wave = threadIdx.x >> 5;
    const int lane = threadIdx.x & 31;
    const int half = lane >> 4;   // 0: lanes 0-15, 1: lanes 16-31
    const int l16  = lane & 15;
    const int row_base = (blockIdx.x * WAVES + wave) * 16;

    // Stage 16x128 tile of X into LDS, fully coalesced b128 loads.
    #pragma unroll
    for (int it = 0; it < 16; ++it) {
        int f = (it * 32 + lane) * 4;      // float offset in tile
        int r = f >> 7;
        int c = f & 127;
        int gr = row_base + r;
        if (gr > N - 1) gr = N - 1;        // clamp tail reads
        const float4 v = *(const float4*)(X + (size_t)gr * DIN + c);
        float* dst = &xs[wave][r * XS_STRIDE + c];
        dst[0] = v.x; dst[1] = v.y; dst[2] = v.z; dst[3] = v.w;
    }
    __syncthreads();

    v8f acc[4] = {v8f{}, v8f{}, v8f{}, v8f{}};

    // K loop: 128 / 4 = 32 WMMA steps per column tile.
    for (int k0 = 0; k0 < DIN; k0 += 4) {
        const int ka = k0 + 2 * half;      // A layout: upper lane half holds K+2
        v2f a;
        a.x = xs[wave][l16 * XS_STRIDE + ka];
        a.y = xs[wave][l16 * XS_STRIDE + ka + 1];
        const float* w0 = W + (size_t)ka * DOUT + l16;   // row ka
        const float* w1 = w0 + DOUT;                      // row ka+1
        #pragma unroll
        for (int t = 0; t < 4; ++t) {
            v2f b;
            b.x = w0[t * 16];
            b.y = w1[t * 16];
            acc[t] = __builtin_amdgcn_wmma_f32_16x16x4_f32(
                false, a, false, b, (short)0, acc[t], false, false);
        }
    }

    // Epilogue: add bias, accumulate per-lane partials of H@a_src / H@a_dst.
    float psS[8], psD[8];
    #pragma unroll
    for (int v = 0; v < 8; ++v) { psS[v] = 0.0f; psD[v] = 0.0f; }

    #pragma unroll
    for (int t = 0; t < 4; ++t) {
        const int n  = t * 16 + l16;
        const float bn = bias[n];
        const float as = a_src[n];
        const float ad = a_dst[n];
        #pragma unroll
        for (int v = 0; v < 8; ++v) {
            float h = acc[t][v] + bn;
            acc[t][v] = h;
            psS[v] += h * as;
            psD[v] += h * ad;
        }
    }

    // Store H. C/D layout: VGPR v holds row v (lanes 0-15) / row v+8 (lanes 16-31).
    #pragma unroll
    for (int t = 0; t < 4; ++t) {
        #pragma unroll
        for (int v = 0; v < 8; ++v) {
            int row = row_base + v + 8 * half;
            if (row < N) H[(size_t)row * DOUT + t * 16 + l16] = acc[t][v];
        }
    }

    // Reduce dot-product partials across the 16 lanes of each half (wave32 shfl).
    #pragma unroll
    for (int v = 0; v < 8; ++v) {
        #pragma unroll
        for (int off = 1; off < 16; off <<= 1) {
            psS[v] += __shfl_xor(psS[v], off, 32);
            psD[v] += __shfl_xor(psD[v], off, 32);
        }
    }
    if (l16 == 0) {
        #pragma unroll
        for (int v = 0; v < 8; ++v) {
            int row = row_base + v + 8 * half;
            if (row < N) { ssrc[row] = psS[v]; sdst[row] = psD[v]; }
        }
    }
}

// ---------------------------------------------------------------------------
// Kernel 2: per-edge ELU score + segment max over destination (encoded atomicMax)
// ---------------------------------------------------------------------------
__global__ void edge_max(const float* __restrict__ ssrc, const float* __restrict__ sdst,
                         const int* __restrict__ esrc, const int* __restrict__ edst,
                         unsigned* __restrict__ menc, int E)
{
    int i = blockIdx.x * blockDim.x + threadIdx.x;
    if (i >= E) return;
    int d = edst[i];
    float e = elu1(ssrc[esrc[i]] + sdst[d]);
    atomicMax(&menc[d], enc_f32(e));
}

// ---------------------------------------------------------------------------
// Kernel 3: ex = exp(e - m[dst]); store numerator; atomic segment sum of denom
// ---------------------------------------------------------------------------
__global__ void edge_exp(const float* __restrict__ ssrc, const float* __restrict__ sdst,
                         const int* __restrict__ esrc, const int* __restrict__ edst,
                         const unsigned* __restrict__ menc,
                         float* __restrict__ denom, float* __restrict__ wgt, int E)
{
    int i = blockIdx.x * blockDim.x + threadIdx.x;
    if (i >= E) return;
    int d = edst[i];
    float e  = elu1(ssrc[esrc[i]] + sdst[d]);
    float ex = expf(e - dec_f32(menc[d]));
    wgt[i] = ex;
    unsafeAtomicAdd(&denom[d], ex);
}

// ---------------------------------------------------------------------------
// Kernel 4: normalize edge weights
// ---------------------------------------------------------------------------
__global__ void edge_norm(const int* __restrict__ edst, const float* __restrict__ denom,
                          float* __restrict__ wgt, int E)
{
    int i = blockIdx.x * blockDim.x + threadIdx.x;
    if (i >= E) return;
    float dn = denom[edst[i]];
    wgt[i] = wgt[i] / (dn > 0.0f ? dn : 1.0f);
}

// ---------------------------------------------------------------------------
// Kernel 5: out[dst] += H[src] * w  — one thread per (edge, column)
// ---------------------------------------------------------------------------
__global__ void aggregate(const float* __restrict__ H, const float* __restrict__ wgt,
                          const int* __restrict__ esrc, const int* __restrict__ edst,
                          float* __restrict__ out, int E)
{
    long long g = (long long)blockIdx.x * blockDim.x + threadIdx.x;
    long long edge = g >> 6;
    int col = (int)(g & 63);
    if (edge >= E) return;
    int s = esrc[edge];
    int d = edst[edge];
    float v = H[(size_t)s * DOUT + col] * wgt[edge];
    unsafeAtomicAdd(&out[(size_t)d * DOUT + col], v);
}

// ---------------------------------------------------------------------------
// Kernel 6: elementwise ELU on the output
// ---------------------------------------------------------------------------
__global__ void elu_out(float* __restrict__ out, long long M)
{
    long long i = (long long)blockIdx.x * blockDim.x + threadIdx.x;
    if (i >= M) return;
    out[i] = elu1(out[i]);
}

extern "C" void kernel_launch(void* const* d_in, const int* in_sizes, int n_in,
                              void* d_out, int out_size, void* d_ws, size_t ws_size,
                              hipStream_t stream) {
    const float* X     = (const float*)d_in[0];
    const int*   esrc  = (const int*)d_in[1];
    const int*   edst  = (const int*)d_in[2];
    const float* W     = (const float*)d_in[3];
    const float* bias  = (const float*)d_in[4];
    const float* a_src = (const float*)d_in[5];
    const float* a_dst = (const float*)d_in[6];
    const int N = in_sizes[0] / DIN;
    const int E = in_sizes[1];
    float* out = (float*)d_out;

    // Workspace carve-up (~33.6 MB for N=100K, E=1.6M)
    char* ws = (char*)d_ws;
    float*    H    = (float*)ws;    ws += (size_t)N * DOUT * sizeof(float);
    float*    ssrc = (float*)ws;    ws += (size_t)N * sizeof(float);
    float*    sdst = (float*)ws;    ws += (size_t)N * sizeof(float);
    unsigned* menc = (unsigned*)ws; ws += (size_t)N * sizeof(unsigned);
    float*    dnm  = (float*)ws;    ws += (size_t)N * sizeof(float);
    float*    wgt  = (float*)ws;    ws += (size_t)E * sizeof(float);

    hipMemsetAsync(menc, 0, (size_t)N * sizeof(unsigned), stream); // enc(-NaN) floor
    hipMemsetAsync(dnm,  0, (size_t)N * sizeof(float), stream);
    hipMemsetAsync(out,  0, (size_t)N * DOUT * sizeof(float), stream);

    const int rows_per_block = 16 * WAVES;
    gat_gemm<<<(N + rows_per_block - 1) / rows_per_block, 32 * WAVES, 0, stream>>>(
        X, W, bias, a_src, a_dst, H, ssrc, sdst, N);

    int eb = (E + 255) / 256;
    edge_max <<<eb, 256, 0, stream>>>(ssrc, sdst, esrc, edst, menc, E);
    edge_exp <<<eb, 256, 0, stream>>>(ssrc, sdst, esrc, edst, menc, dnm, wgt, E);
    edge_norm<<<eb, 256, 0, stream>>>(edst, dnm, wgt, E);

    long long total = (long long)E * DOUT;
    aggregate<<<(unsigned)((total + 255) / 256), 256, 0, stream>>>(H, wgt, esrc, edst, out, E);

    long long M = (long long)N * DOUT;
    elu_out<<<(unsigned)((M + 255) / 256), 256, 0, stream>>>(out, M);
}